// cmdnet3_36936718745656
// MI455X (gfx1250) — compile-verified
//
#include <hip/hip_runtime.h>
#include <hip/hip_bf16.h>

typedef __attribute__((ext_vector_type(2))) float v2f;
typedef __attribute__((ext_vector_type(8))) float v8f;

#define WAVES_PER_BLOCK 8
#define K_TX 32
#define N_RX 64
#define NUM_ITER 64
#define LDS_STRIDE 36   // floats; 36*4=144 bytes => every row 16B aligned, conflict-free column reads

__device__ __forceinline__ float bcast_lane(float v, int srclane) {
    return __int_as_float(__builtin_amdgcn_readlane(__float_as_int(v), srclane));
}

__global__ __launch_bounds__(256, 1) void cmdnet_kernel(
    const float* __restrict__ yt,    // [B,64]
    const float* __restrict__ Ht,    // [B,64,32] row-major (n_rx major)
    const float* __restrict__ sig0,  // [B]
    const float* __restrict__ taui,  // [NUM_ITER+1]
    const float* __restrict__ delta, // [NUM_ITER]
    float* __restrict__ out_ft,      // [B,32,2]
    float* __restrict__ out_xt,      // [B,32]
    int B)
{
    __shared__ __align__(16) float lds_hh[WAVES_PER_BLOCK][K_TX * LDS_STRIDE];

    const int wave = threadIdx.x >> 5;
    const int lane = threadIdx.x & 31;
    const int b = blockIdx.x * WAVES_PER_BLOCK + wave;
    if (b >= B) return;   // wave-uniform: EXEC stays all-1s for WMMA

    const float* Hb = Ht + (size_t)b * (N_RX * K_TX);
    const float* yb = yt + (size_t)b * N_RX;

    const int mlo   = lane & 15;        // M / N index within 16-wide tile
    const int khalf = (lane >> 4) * 2;  // K sub-offset: lanes 0-15 -> K{0,1}, 16-31 -> K{2,3}

    // ---- Phase 1: HH = H^T H via fp32 WMMA (16x16x4), yH folded into same pass ----
    // Tile T_c (c = column half of H): per lane, VGPR p holds H[kc*4 + khalf + p, c*16 + mlo].
    // This register layout is simultaneously a valid A (16x4) and B (4x16) operand,
    // so each tile is loaded once and reused for both operand roles (HH is symmetric).
    v8f acc00 = {}, acc01 = {}, acc10 = {}, acc11 = {};
    float yacc0 = 0.f, yacc1 = 0.f;

    #pragma unroll
    for (int kc = 0; kc < 16; ++kc) {
        const int r0 = kc * 4 + khalf;
        v2f t0, t1;
        t0.x = Hb[(size_t)r0 * K_TX + mlo];
        t0.y = Hb[(size_t)(r0 + 1) * K_TX + mlo];
        t1.x = Hb[(size_t)r0 * K_TX + 16 + mlo];
        t1.y = Hb[(size_t)(r0 + 1) * K_TX + 16 + mlo];
        const float y0 = yb[r0], y1 = yb[r0 + 1];
        yacc0 = fmaf(y0, t0.x, fmaf(y1, t0.y, yacc0));
        yacc1 = fmaf(y0, t1.x, fmaf(y1, t1.y, yacc1));
        // D = A*B + C   (8 args: neg_a, A, neg_b, B, c_mod, C, reuse_a, reuse_b)
        acc00 = __builtin_amdgcn_wmma_f32_16x16x4_f32(false, t0, false, t0, (short)0, acc00, false, false);
        acc01 = __builtin_amdgcn_wmma_f32_16x16x4_f32(false, t0, false, t1, (short)0, acc01, false, false);
        acc10 = __builtin_amdgcn_wmma_f32_16x16x4_f32(false, t1, false, t0, (short)0, acc10, false, false);
        acc11 = __builtin_amdgcn_wmma_f32_16x16x4_f32(false, t1, false, t1, (short)0, acc11, false, false);
    }

    // yH[c*16 + mlo] split across lane halves -> xor-16 reduce; then lane k's yH is
    // yacc0 (k<16) or yacc1 (k>=16) in its own lane.
    yacc0 += __shfl_xor(yacc0, 16, 32);
    yacc1 += __shfl_xor(yacc1, 16, 32);
    const float yh = (lane < 16) ? yacc0 : yacc1;

    // ---- Phase 2: transpose HH from WMMA C/D layout to "lane k holds row k" via LDS ----
    // C/D layout: VGPR r, lane l  ->  row = i*16 + (l>>4)*8 + r, col = j*16 + (l&15)
    float* lhh = lds_hh[wave];
    const int rbase = (lane >> 4) * 8;
    #pragma unroll
    for (int r = 0; r < 8; ++r) {
        lhh[(0 * 16 + rbase + r) * LDS_STRIDE + 0 * 16 + mlo] = acc00[r];
        lhh[(0 * 16 + rbase + r) * LDS_STRIDE + 1 * 16 + mlo] = acc01[r];
        lhh[(1 * 16 + rbase + r) * LDS_STRIDE + 0 * 16 + mlo] = acc10[r];
        lhh[(1 * 16 + rbase + r) * LDS_STRIDE + 1 * 16 + mlo] = acc11[r];
    }
    // Same-wave DS ops are in-order; wait for stores + stop compiler reordering.
    asm volatile("s_wait_dscnt 0x0" ::: "memory");

    float hh[K_TX];
    const float4* rowp = (const float4*)(lhh + lane * LDS_STRIDE); // 144B stride -> 16B aligned
    #pragma unroll
    for (int q = 0; q < 8; ++q) {
        float4 v = rowp[q];
        hh[q * 4 + 0] = v.x; hh[q * 4 + 1] = v.y;
        hh[q * 4 + 2] = v.z; hh[q * 4 + 3] = v.w;
    }

    // ---- Phase 3: 64 CMDNet iterations, all state in VGPRs ----
    const float s0 = sig0[b];
    const float sig2 = s0 * s0;
    const float log_alpha = -0.6931471805599453f; // log(0.5), equal for both symbols
    float tau = fabsf(taui[0]);
    float g0 = 0.f, g1 = 0.f;
    float f0, f1, xt;

    auto recompute = [&]() {
        float a0 = (log_alpha + g0) * tau;
        float a1 = (log_alpha + g1) * tau;
        float mx = fmaxf(a0, a1);
        float e0 = __expf(a0 - mx);
        float e1 = __expf(a1 - mx);
        float inv = __frcp_rn(e0 + e1);
        f0 = e0 * inv;
        f1 = e1 * inv;
        xt = f1 - f0;   // m = [-1, +1]
    };
    recompute();

    for (int it = 0; it < NUM_ITER; ++it) {
        // xHH[k] = sum_j xt[j] * HH[j,k]; HH symmetric => use row k held in this lane.
        // 4 independent partial sums break the FMA dependency chain (ILP / dual-issue).
        float p0 = 0.f, p1 = 0.f, p2 = 0.f, p3 = 0.f;
        #pragma unroll
        for (int j = 0; j < K_TX; j += 4) {
            p0 = fmaf(bcast_lane(xt, j + 0), hh[j + 0], p0);
            p1 = fmaf(bcast_lane(xt, j + 1), hh[j + 1], p1);
            p2 = fmaf(bcast_lane(xt, j + 2), hh[j + 2], p2);
            p3 = fmaf(bcast_lane(xt, j + 3), hh[j + 3], p3);
        }
        const float xhh = (p0 + p1) + (p2 + p3);
        const float r = xhh - yh;
        // grad_x_m = tau * ft_m * (m_m - xt);  grad_L = sig2*(1-exp(-G)) + grad_x * r
        const float taur = tau * r;
        const float em0 = __expf(-g0);           // independent transcendental streams
        const float em1 = __expf(-g1);
        const float gl0 = fmaf(f0 * (-1.f - xt), taur, sig2 * (1.f - em0));
        const float gl1 = fmaf(f1 * ( 1.f - xt), taur, sig2 * (1.f - em1));
        const float d = delta[it];
        g0 = fmaf(-d, gl0, g0);
        g1 = fmaf(-d, gl1, g1);
        tau = fabsf(taui[it + 1]);
        recompute();
    }

    // ---- Output: ft [B,32,2] then xt [B,32] ----
    float2 f = make_float2(f0, f1);
    *(float2*)(out_ft + (size_t)b * (K_TX * 2) + lane * 2) = f;
    out_xt[(size_t)b * K_TX + lane] = xt;
}

extern "C" void kernel_launch(void* const* d_in, const int* in_sizes, int n_in,
                              void* d_out, int out_size, void* d_ws, size_t ws_size,
                              hipStream_t stream) {
    const float* yt    = (const float*)d_in[0];
    const float* Ht    = (const float*)d_in[1];
    const float* sig0  = (const float*)d_in[2];
    const float* taui  = (const float*)d_in[3];
    const float* delta = (const float*)d_in[4];
    const int B = in_sizes[2];                 // sigmat0 has B elements

    float* out_ft = (float*)d_out;                        // [B,32,2]
    float* out_xt = (float*)d_out + (size_t)B * K_TX * 2; // [B,32]

    const int blocks = (B + WAVES_PER_BLOCK - 1) / WAVES_PER_BLOCK;
    cmdnet_kernel<<<blocks, WAVES_PER_BLOCK * 32, 0, stream>>>(
        yt, Ht, sig0, taui, delta, out_ft, out_xt, B);
}